// DCNv4_7876970021406
// MI455X (gfx1250) — compile-verified
//
#include <hip/hip_runtime.h>

// ---------------------------------------------------------------------------
// DCNv4 for MI455X (gfx1250).
// fp32 GEMMs on V_WMMA_F32_16X16X4_F32 (fp32-exact accumulate, matches the
// fp32 reference).  Register-blocked wave tiles (TM x TN 16x16 subtiles),
// address-space(1) pointers + compile-time strides so all fragment loads are
// global_load_* with immediate offsets (no flat path, minimal pointer VALU).
// ---------------------------------------------------------------------------

typedef __attribute__((ext_vector_type(2))) float v2f;
typedef __attribute__((ext_vector_type(4))) float v4f;
typedef __attribute__((ext_vector_type(8))) float v8f;

// global (device memory) address space pointers -> force global_load/store
typedef const __attribute__((address_space(1))) float* gcfp;
typedef       __attribute__((address_space(1))) float* gfp;
typedef const __attribute__((address_space(1))) v2f*   gcv2p;
typedef const __attribute__((address_space(1))) v4f*   gcv4p;

#define HH_   56
#define WW_   56
#define HW_   3136          // 56*56 ; divisible by 32 -> 2-tile M groups stay in-batch
#define C_    256
#define OMD_  432           // G*P*3 = 16*9*3
#define G_    16
#define GC_   16
#define MTOT_ 25088         // 8*56*56

// ---------------------------------------------------------------------------
// Row-major-output GEMM:  O[m, n] = sum_k A[m,k] * W[n,k] + bias[n]
// A is column-major with compile-time leading dim LDA, batched in chunks of
// CHUNK rows with BS floats between batches.  One wave per
// (TM*16) x (TN*16) output tile; one A pointer + one B pointer per wave,
// all subtile accesses via immediate offsets.
// ---------------------------------------------------------------------------
template <int TM, int TN, int LDA, int LDO, int CHUNK, long long BS>
__global__ __launch_bounds__(32) void gemm_rm(
    const float* __restrict__ A_, const float* __restrict__ W_,
    const float* __restrict__ Bias_, float* __restrict__ O_)
{
    gcfp A    = (gcfp)A_;
    gcfp W    = (gcfp)W_;
    gcfp Bias = (gcfp)Bias_;
    gfp  O    = (gfp)O_;

    const int m0   = blockIdx.x * (TM * 16);
    const int n0   = blockIdx.y * (TN * 16);
    const int lane = threadIdx.x;
    const int half = lane >> 4;          // 0: K = 0,1   1: K = 2,3
    const int lid  = lane & 15;

    const int nb = m0 / CHUNK;
    const int ml = m0 % CHUNK;

    gcfp aptr = A + (size_t)nb * BS + ml + lid + (size_t)(2 * half) * LDA;
    gcfp bptr = W + (size_t)(n0 + lid) * C_ + 2 * half;

    v8f acc[TM][TN];
#pragma unroll
    for (int t = 0; t < TM; ++t)
#pragma unroll
        for (int u = 0; u < TN; ++u)
            acc[t][u] = (v8f){0.f, 0.f, 0.f, 0.f, 0.f, 0.f, 0.f, 0.f};

#pragma unroll 2
    for (int k = 0; k < C_; k += 4) {
        v2f a[TM];
#pragma unroll
        for (int t = 0; t < TM; ++t) {
            a[t][0] = aptr[t * 16];           // imm offset
            a[t][1] = aptr[t * 16 + LDA];     // imm offset (LDA compile-time)
        }
        v2f b[TN];
#pragma unroll
        for (int u = 0; u < TN; ++u)
            b[u] = *(gcv2p)(bptr + u * (16 * C_));   // imm offset 16KB * u

#pragma unroll
        for (int t = 0; t < TM; ++t)
#pragma unroll
            for (int u = 0; u < TN; ++u)
                acc[t][u] = __builtin_amdgcn_wmma_f32_16x16x4_f32(
                    false, a[t], false, b[u], (short)0, acc[t][u], false, false);

        aptr += 4 * (size_t)LDA;
        bptr += 4;
    }

    // D layout: VGPR r -> M = r + 8*half, N = lid.  Row-major coalesced stores.
#pragma unroll
    for (int u = 0; u < TN; ++u) {
        const float bias = Bias[n0 + u * 16 + lid];
#pragma unroll
        for (int t = 0; t < TM; ++t) {
            gfp orow = O + (size_t)(m0 + t * 16 + 8 * half) * LDO + n0 + u * 16 + lid;
#pragma unroll
            for (int r = 0; r < 8; ++r)
                orow[r * LDO] = acc[t][u][r] + bias;   // imm offsets
        }
    }
}

// ---------------------------------------------------------------------------
// NCHW-output GEMM (final projection):
//   out[nb, n, h, w] = sum_k core_cm[k*M + m] * W[n,k] + bias[n]
// A column-major (LDA = MTOT_) -> coalesced loads.  Each 16x16 subtile is
// transposed through padded LDS so stores are 64B-contiguous per channel.
// ---------------------------------------------------------------------------
template <int TM, int TN>
__global__ __launch_bounds__(32) void gemm_nchw(
    const float* __restrict__ A_, const float* __restrict__ W_,
    const float* __restrict__ Bias_, float* __restrict__ out_)
{
    __shared__ float tile[TM][TN][16][17];   // +1 pad: conflict-free transpose

    gcfp A    = (gcfp)A_;
    gcfp W    = (gcfp)W_;
    gcfp Bias = (gcfp)Bias_;
    gfp  out  = (gfp)out_;

    const int m0   = blockIdx.x * (TM * 16);
    const int n0   = blockIdx.y * (TN * 16);
    const int lane = threadIdx.x;
    const int half = lane >> 4;
    const int lid  = lane & 15;

    gcfp aptr = A + m0 + lid + (size_t)(2 * half) * MTOT_;
    gcfp bptr = W + (size_t)(n0 + lid) * C_ + 2 * half;

    v8f acc[TM][TN];
#pragma unroll
    for (int t = 0; t < TM; ++t)
#pragma unroll
        for (int u = 0; u < TN; ++u)
            acc[t][u] = (v8f){0.f, 0.f, 0.f, 0.f, 0.f, 0.f, 0.f, 0.f};

#pragma unroll 2
    for (int k = 0; k < C_; k += 4) {
        v2f a[TM];
#pragma unroll
        for (int t = 0; t < TM; ++t) {
            a[t][0] = aptr[t * 16];
            a[t][1] = aptr[t * 16 + MTOT_];
        }
        v2f b[TN];
#pragma unroll
        for (int u = 0; u < TN; ++u)
            b[u] = *(gcv2p)(bptr + u * (16 * C_));

#pragma unroll
        for (int t = 0; t < TM; ++t)
#pragma unroll
            for (int u = 0; u < TN; ++u)
                acc[t][u] = __builtin_amdgcn_wmma_f32_16x16x4_f32(
                    false, a[t], false, b[u], (short)0, acc[t][u], false, false);

        aptr += 4 * (size_t)MTOT_;
        bptr += 4;
    }

#pragma unroll
    for (int u = 0; u < TN; ++u) {
        const float bias = Bias[n0 + u * 16 + lid];
#pragma unroll
        for (int t = 0; t < TM; ++t)
#pragma unroll
            for (int r = 0; r < 8; ++r)
                tile[t][u][r + 8 * half][lid] = acc[t][u][r] + bias;  // [M][N]
    }
    __syncthreads();

#pragma unroll
    for (int t = 0; t < TM; ++t) {
        const int mt = m0 + t * 16;
        const int nb = mt / HW_;
        const int ml = mt % HW_;
        gfp obase = out + ((size_t)nb * C_ + n0) * HW_ + ml;
#pragma unroll
        for (int u = 0; u < TN; ++u) {
#pragma unroll
            for (int it = 0; it < 8; ++it) {
                const int cc = it * 2 + half;                  // channel in subtile
                obase[(u * 16 + cc) * HW_ + lid] = tile[t][u][lid][cc];
            }
        }
    }
}

// ---------------------------------------------------------------------------
// Deformable bilinear sampling core.  One thread per (m, g): 9 points x 4 taps
// x 16 channels, float4 streaming loads from val.  Output written COLUMN-major
// (core[c*M + m]) so the final GEMM gets coalesced column loads; stores are
// coalesced across threadIdx.x (consecutive m).
// ---------------------------------------------------------------------------
__global__ __launch_bounds__(256) void dcnv4_core_kernel(
    const float* __restrict__ val_,
    const float* __restrict__ om_,
    float* __restrict__ core_)
{
    gcfp val  = (gcfp)val_;
    gcfp om   = (gcfp)om_;
    gfp  core = (gfp)core_;

    const int m = blockIdx.x * 256 + threadIdx.x;
    const int g = blockIdx.y;

    const int nb  = m / HW_;
    const int rem = m % HW_;
    const int h   = rem / WW_;
    const int w   = rem % WW_;

    // per-(m,g) packet: [dx0,dy0,...,dx8,dy8, m0..m8]
    gcfp omg = om + (size_t)m * OMD_ + g * 27;
    float pkt[27];
#pragma unroll
    for (int i = 0; i < 27; ++i) pkt[i] = omg[i];

    float acc[GC_];
#pragma unroll
    for (int c = 0; c < GC_; ++c) acc[c] = 0.f;

    gcfp vbase = val + (size_t)nb * HW_ * C_ + g * GC_;

#pragma unroll
    for (int p = 0; p < 9; ++p) {
        const int ii = p / 3, jj = p % 3;                  // meshgrid 'ij'
        const float loc_h = (float)(h - 1 + ii) + pkt[2 * p + 1];
        const float loc_w = (float)(w - 1 + jj) + pkt[2 * p + 0];
        const float h0f = floorf(loc_h), w0f = floorf(loc_w);
        const float fh = loc_h - h0f, fw = loc_w - w0f;
        const int h0 = (int)h0f, w0 = (int)w0f;
        const float mk = pkt[18 + p];
        const float wt4[4] = { (1.f - fh) * (1.f - fw) * mk,
                               (1.f - fh) * fw * mk,
                               fh * (1.f - fw) * mk,
                               fh * fw * mk };
#pragma unroll
        for (int q = 0; q < 4; ++q) {
            const int hs = h0 + (q >> 1);
            const int ws = w0 + (q & 1);
            if (hs >= 0 && hs < HH_ && ws >= 0 && ws < WW_) {
                const float wt = wt4[q];
                gcfp s = vbase + ((size_t)hs * WW_ + ws) * C_;   // 64B aligned
                const v4f s0 = *(gcv4p)(s);
                const v4f s1 = *(gcv4p)(s + 4);
                const v4f s2 = *(gcv4p)(s + 8);
                const v4f s3 = *(gcv4p)(s + 12);
#pragma unroll
                for (int j = 0; j < 4; ++j) {
                    acc[j]      += wt * s0[j];
                    acc[4 + j]  += wt * s1[j];
                    acc[8 + j]  += wt * s2[j];
                    acc[12 + j] += wt * s3[j];
                }
            }
        }
    }

    // column-major store: coalesced over threadIdx.x for each channel
#pragma unroll
    for (int c = 0; c < GC_; ++c)
        core[(size_t)(g * GC_ + c) * MTOT_ + m] = acc[c];
}

// ---------------------------------------------------------------------------
extern "C" void kernel_launch(void* const* d_in, const int* in_sizes, int n_in,
                              void* d_out, int out_size, void* d_ws, size_t ws_size,
                              hipStream_t stream)
{
    (void)in_sizes; (void)n_in; (void)out_size; (void)ws_size;

    const float* x   = (const float*)d_in[0];
    const float* vw  = (const float*)d_in[1];
    const float* vb  = (const float*)d_in[2];
    const float* omw = (const float*)d_in[3];
    const float* omb = (const float*)d_in[4];
    const float* ow  = (const float*)d_in[5];
    const float* ob  = (const float*)d_in[6];
    float* out = (float*)d_out;

    // workspace: val (M*256) | om (M*432) | core column-major (256*M)
    float* val  = (float*)d_ws;
    float* om   = val + (size_t)MTOT_ * C_;
    float* core = om  + (size_t)MTOT_ * OMD_;

    // value projection: 25088 x 256 <- xp @ value_w^T
    dim3 gv(MTOT_ / 32, C_ / 64);                  // 784 x 4
    gemm_rm<2, 4, HW_, C_, HW_, (long long)C_ * HW_>
        <<<gv, 32, 0, stream>>>(x, vw, vb, val);

    // offset/mask projection: 25088 x 432
    dim3 go(MTOT_ / 32, OMD_ / 48);                // 784 x 9
    gemm_rm<2, 3, HW_, OMD_, HW_, (long long)C_ * HW_>
        <<<go, 32, 0, stream>>>(x, omw, omb, om);

    // deformable sampling core
    dim3 gc(MTOT_ / 256, G_);                      // 98 x 16
    dcnv4_core_kernel<<<gc, 256, 0, stream>>>(val, om, core);

    // output projection -> NCHW
    dim3 gf(MTOT_ / 32, C_ / 64);                  // 784 x 4
    gemm_nchw<2, 4><<<gf, 32, 0, stream>>>(core, ow, ob, out);
}